// SSDGhost_73083163509056
// MI455X (gfx1250) — compile-verified
//
#include <hip/hip_runtime.h>
#include <math.h>

typedef __attribute__((ext_vector_type(2))) float v2f;
typedef __attribute__((ext_vector_type(8))) float v8f;

#define NCLS    21
#define NTOT    24512
#define KTOP    100
#define SCORE_T 0.05f
#define NMS_T   0.45f
#define BN_EPS  1e-5f

// ---------------------------------------------------------------------------
// WMMA f32 16x16x4 GEMM core, A staged in LDS, 4 N-tiles per wave.
// A frag (16x4):  lane n(0..15) -> M=n ; VGPR v -> K = 2*half + v
// B frag (4x16):  VGPR v -> K = 2*half + v ; col N = lane&15
// C/D  (16x16):   VGPR v -> M = v + 8*half ; N = lane&15
// ---------------------------------------------------------------------------
template <int CIN>
__device__ __forceinline__ void gemm_core(const float* __restrict__ Xb,
                                          const float* __restrict__ Wl,  // LDS, padded rows
                                          int HW, int co0, const int pj[4],
                                          int lane, v8f acc[4]) {
  const int half = lane >> 4;
  const int nn   = lane & 15;
  const int arow = (co0 + nn) * CIN;
  for (int k = 0; k < CIN; k += 4) {
    v2f a;
    a.x = Wl[arow + k + 2 * half];
    a.y = Wl[arow + k + 2 * half + 1];
    const int xrow = (k + 2 * half) * HW + nn;
#pragma unroll
    for (int j = 0; j < 4; ++j) {
      v2f b;
      b.x = Xb[xrow + pj[j]];
      b.y = Xb[xrow + HW + pj[j]];
      acc[j] = __builtin_amdgcn_wmma_f32_16x16x4_f32(false, a, false, b,
                                                     (short)0, acc[j], false, false);
    }
  }
}

// Common wave->tile mapping: returns false if wave has no work.
__device__ __forceinline__ bool tile_map(int Cout, int HW, int B, int& b, int& co0,
                                         int& pixT, int& pt4, int pj[4]) {
  int wave = (int)((blockIdx.x * blockDim.x + threadIdx.x) >> 5);
  int coT  = (Cout + 15) >> 4;
  pixT     = HW >> 4;
  int pixG = (pixT + 3) >> 2;
  int total = coT * pixG * B;
  if (wave >= total) return false;
  b       = wave / (coT * pixG);
  int rem = wave % (coT * pixG);
  co0     = (rem / pixG) * 16;
  pt4     = (rem % pixG) * 4;
#pragma unroll
  for (int j = 0; j < 4; ++j) {
    int pt = pt4 + j;
    if (pt > pixT - 1) pt = pixT - 1;   // clamp: redundant compute, store guarded
    pj[j] = pt * 16;
  }
  return true;
}

// Cooperative LDS weight staging, zero-padded to 16-row multiple.
template <int CIN, int ROWS>
__device__ __forceinline__ void stage_weights(float* Wl, const float* __restrict__ Wg,
                                              int Cout) {
  int rows = ((Cout + 15) >> 4) << 4;
  for (int i = threadIdx.x; i < rows * CIN; i += 256) {
    int r = i / CIN, cidx = i - r * CIN;
    Wl[i] = (r < Cout) ? Wg[r * CIN + cidx] : 0.f;
  }
}

// ---- 1x1 conv, plain NCHW output (+bias) ----------------------------------
template <int CIN>
__global__ void k_gemm_plain_t(const float* __restrict__ X, const float* __restrict__ Wg,
                               const float* __restrict__ bias, float* __restrict__ Y,
                               int Cout, int HW, int B) {
  __shared__ float Wl[64 * CIN];
  stage_weights<CIN, 64>(Wl, Wg, Cout);
  __syncthreads();
  int b, co0, pixT, pt4, pj[4];
  if (!tile_map(Cout, HW, B, b, co0, pixT, pt4, pj)) return;
  const float* Xb = X + (size_t)b * CIN * HW;
  v8f acc[4];
#pragma unroll
  for (int j = 0; j < 4; ++j) acc[j] = (v8f){0.f,0.f,0.f,0.f,0.f,0.f,0.f,0.f};
  int lane = threadIdx.x & 31;
  gemm_core<CIN>(Xb, Wl, HW, co0, pj, lane, acc);
  int half = lane >> 4, nn = lane & 15;
  float* Yb = Y + (size_t)b * Cout * HW;
#pragma unroll
  for (int j = 0; j < 4; ++j) {
    if (pt4 + j >= pixT) break;
#pragma unroll
    for (int v = 0; v < 8; ++v) {
      int co = co0 + v + 8 * half;
      Yb[co * HW + pj[j] + nn] = acc[j][v] + bias[co];
    }
  }
}

// ---- 1x1 conv + BN + ReLU (pointwise of dws blocks) ------------------------
__global__ void k_gemm_pw_bnrelu(const float* __restrict__ X, const float* __restrict__ Wg,
                                 const float* __restrict__ g, const float* __restrict__ beta,
                                 const float* __restrict__ mean, const float* __restrict__ var,
                                 float* __restrict__ Y, int HW, int B) {
  const int CIN = 64, Cout = 64;
  __shared__ float Wl[64 * 64];
  stage_weights<64, 64>(Wl, Wg, Cout);
  __syncthreads();
  int b, co0, pixT, pt4, pj[4];
  if (!tile_map(Cout, HW, B, b, co0, pixT, pt4, pj)) return;
  const float* Xb = X + (size_t)b * CIN * HW;
  v8f acc[4];
#pragma unroll
  for (int j = 0; j < 4; ++j) acc[j] = (v8f){0.f,0.f,0.f,0.f,0.f,0.f,0.f,0.f};
  int lane = threadIdx.x & 31;
  gemm_core<64>(Xb, Wl, HW, co0, pj, lane, acc);
  int half = lane >> 4, nn = lane & 15;
  float* Yb = Y + (size_t)b * Cout * HW;
#pragma unroll
  for (int j = 0; j < 4; ++j) {
    if (pt4 + j >= pixT) break;
#pragma unroll
    for (int v = 0; v < 8; ++v) {
      int co = co0 + v + 8 * half;
      float sc  = g[co] * rsqrtf(var[co] + BN_EPS);
      float val = (acc[j][v] - mean[co]) * sc + beta[co];
      Yb[co * HW + pj[j] + nn] = fmaxf(val, 0.f);
    }
  }
}

// ---- head 1x1 conv + bias, scattered into [B, NTOT, d] layout --------------
template <int ROWS>
__global__ void k_gemm_head_t(const float* __restrict__ X, const float* __restrict__ Wg,
                              const float* __restrict__ bias, float* __restrict__ out,
                              int Cout, int HW, int B, int base, int na, int d) {
  __shared__ float Wl[ROWS * 64];
  stage_weights<64, ROWS>(Wl, Wg, Cout);
  __syncthreads();
  int b, co0, pixT, pt4, pj[4];
  if (!tile_map(Cout, HW, B, b, co0, pixT, pt4, pj)) return;
  const float* Xb = X + (size_t)b * 64 * HW;
  v8f acc[4];
#pragma unroll
  for (int j = 0; j < 4; ++j) acc[j] = (v8f){0.f,0.f,0.f,0.f,0.f,0.f,0.f,0.f};
  int lane = threadIdx.x & 31;
  gemm_core<64>(Xb, Wl, HW, co0, pj, lane, acc);
  int half = lane >> 4, nn = lane & 15;
  float* ob = out + (size_t)b * NTOT * d;
#pragma unroll
  for (int j = 0; j < 4; ++j) {
    if (pt4 + j >= pixT) break;
#pragma unroll
    for (int v = 0; v < 8; ++v) {
      int co = co0 + v + 8 * half;
      if (co < Cout) {
        int a = co / d, jj = co - a * d;
        int nidx = base + (pj[j] + nn) * na + a;
        ob[nidx * d + jj] = acc[j][v] + bias[co];
      }
    }
  }
}

// ---- dst += up2(src)  (nearest x2) -----------------------------------------
__global__ void k_add_up2(float* __restrict__ dst, const float* __restrict__ src,
                          int C, int H, int W, int B) {
  int i = blockIdx.x * blockDim.x + threadIdx.x;
  int total = B * C * H * W;
  if (i >= total) return;
  int x = i % W, t = i / W;
  int y = t % H; t /= H;
  int c = t % C; int b = t / C;
  dst[i] += src[(((size_t)b * C + c) * (H >> 1) + (y >> 1)) * (W >> 1) + (x >> 1)];
}

// ---- partial conv: 3x3 over first 16 of 64 channels ------------------------
__global__ void k_pconv16(const float* __restrict__ P, const float* __restrict__ Wc,
                          float* __restrict__ Tmp, int H, int W, int B) {
  int i = blockIdx.x * blockDim.x + threadIdx.x;
  int total = B * 16 * H * W;
  if (i >= total) return;
  int x = i % W, t = i / W;
  int y = t % H; t /= H;
  int co = t % 16; int b = t / 16;
  float acc = 0.f;
  for (int ci = 0; ci < 16; ++ci) {
    const float* ip = P + ((size_t)b * 64 + ci) * H * W;
#pragma unroll
    for (int dy = 0; dy < 3; ++dy) {
      int iy = y + dy - 1;
      if ((unsigned)iy >= (unsigned)H) continue;
#pragma unroll
      for (int dx = 0; dx < 3; ++dx) {
        int ix = x + dx - 1;
        if ((unsigned)ix >= (unsigned)W) continue;
        acc = fmaf(Wc[((co * 16 + ci) * 3 + dy) * 3 + dx], ip[iy * W + ix], acc);
      }
    }
  }
  Tmp[i] = acc;
}

__global__ void k_copy16(float* __restrict__ P, const float* __restrict__ Tmp,
                         int H, int W, int B) {
  int i = blockIdx.x * blockDim.x + threadIdx.x;
  int total = B * 16 * H * W;
  if (i >= total) return;
  int hw = i % (H * W), t = i / (H * W);
  int co = t % 16; int b = t / 16;
  P[((size_t)b * 64 + co) * H * W + hw] = Tmp[i];
}

// ---- depthwise 3x3, pad 1, stride s ----------------------------------------
__global__ void k_dwconv(const float* __restrict__ X, const float* __restrict__ Wd,
                         float* __restrict__ Y, int C, int H, int W, int stride, int B) {
  int Ho = (H - 1) / stride + 1, Wo = (W - 1) / stride + 1;
  int i = blockIdx.x * blockDim.x + threadIdx.x;
  int total = B * C * Ho * Wo;
  if (i >= total) return;
  int x = i % Wo, t = i / Wo;
  int y = t % Ho; t /= Ho;
  int c = t % C; int b = t / C;
  float acc = 0.f;
#pragma unroll
  for (int dy = 0; dy < 3; ++dy) {
    int iy = y * stride + dy - 1;
    if ((unsigned)iy >= (unsigned)H) continue;
#pragma unroll
    for (int dx = 0; dx < 3; ++dx) {
      int ix = x * stride + dx - 1;
      if ((unsigned)ix >= (unsigned)W) continue;
      acc = fmaf(Wd[c * 9 + dy * 3 + dx], X[((size_t)(b * C + c) * H + iy) * W + ix], acc);
    }
  }
  Y[((size_t)(b * C + c) * Ho + y) * Wo + x] = acc;
}

// ---- SSD default boxes ------------------------------------------------------
__global__ void k_priors(float* __restrict__ pr) {
  int n = blockIdx.x * blockDim.x + threadIdx.x;
  if (n >= NTOT) return;
  const int fks[5]   = {64, 32, 16, 8, 4};
  const int nas[5]   = {4, 6, 6, 6, 4};
  const int bases[6] = {0, 16384, 22528, 24064, 24448, 24512};
  int k = 0;
  while (n >= bases[k + 1]) k++;
  int loc = n - bases[k];
  int A = nas[k], fk = fks[k];
  int cell = loc / A, a = loc % A;
  int y = cell / fk, x = cell % fk;
  float s  = 0.15f + 0.1875f * k;
  float s1 = (k == 4) ? 1.0f : 0.15f + 0.1875f * (k + 1);
  float sp = sqrtf(s * s1);
  float w, h;
  if      (a == 0) { w = s;  h = s;  }
  else if (a == 1) { w = sp; h = sp; }
  else {
    float r  = ((a - 2) >> 1) == 0 ? 2.0f : 3.0f;
    float sr = sqrtf(r);
    if (((a - 2) & 1) == 0) { w = s * sr; h = s / sr; } else { w = s / sr; h = s * sr; }
  }
  w = fminf(fmaxf(w, 0.f), 1.f); h = fminf(fmaxf(h, 0.f), 1.f);
  float cx = (x + 0.5f) / fk, cy = (y + 0.5f) / fk;
  pr[n * 4 + 0] = fminf(fmaxf(cx - w * 0.5f, 0.f), 1.f);
  pr[n * 4 + 1] = fminf(fmaxf(cy - h * 0.5f, 0.f), 1.f);
  pr[n * 4 + 2] = fminf(fmaxf(cx + w * 0.5f, 0.f), 1.f);
  pr[n * 4 + 3] = fminf(fmaxf(cy + h * 0.5f, 0.f), 1.f);
}

// ---- box decode -------------------------------------------------------------
__global__ void k_decode(const float* __restrict__ reg, const float* __restrict__ pr,
                         float* __restrict__ boxes, int B) {
  int i = blockIdx.x * blockDim.x + threadIdx.x;
  if (i >= B * NTOT) return;
  int n = i % NTOT;
  const float* r = reg + (size_t)i * 4;
  float p0 = pr[n * 4], p1 = pr[n * 4 + 1], p2 = pr[n * 4 + 2], p3 = pr[n * 4 + 3];
  float pw = p2 - p0, ph = p3 - p1;
  float pcx = (p0 + p2) * 0.5f, pcy = (p1 + p3) * 0.5f;
  float cx = r[0] * 0.1f * pw + pcx, cy = r[1] * 0.1f * ph + pcy;
  float w = expf(r[2] * 0.2f) * pw, h = expf(r[3] * 0.2f) * ph;
  float* o = boxes + (size_t)i * 4;
  o[0] = cx - w * 0.5f; o[1] = cy - h * 0.5f; o[2] = cx + w * 0.5f; o[3] = cy + h * 0.5f;
}

// ---- softmax over 21 classes -> probs[B,20,N] -------------------------------
__global__ void k_softmax(const float* __restrict__ cls, float* __restrict__ probs, int B) {
  int i = blockIdx.x * blockDim.x + threadIdx.x;
  if (i >= B * NTOT) return;
  int b = i / NTOT, n = i % NTOT;
  const float* c = cls + (size_t)i * NCLS;
  float mx = c[0];
  for (int j = 1; j < NCLS; ++j) mx = fmaxf(mx, c[j]);
  float e[NCLS], sum = 0.f;
  for (int j = 0; j < NCLS; ++j) { e[j] = expf(c[j] - mx); sum += e[j]; }
  float inv = 1.f / sum;
  for (int cc = 0; cc < NCLS - 1; ++cc)
    probs[((size_t)b * (NCLS - 1) + cc) * NTOT + n] = e[cc + 1] * inv;
}

// ---- per-(image,class) top-100 + greedy NMS ---------------------------------
__global__ void k_nms(const float* __restrict__ boxes, float* __restrict__ probs,
                      float* __restrict__ out) {
  int b = blockIdx.x / (NCLS - 1);
  int c = blockIdx.x % (NCLS - 1);
  float* sc = probs + ((size_t)b * (NCLS - 1) + c) * NTOT;   // destructive (recomputed each call)
  const float* bx = boxes + (size_t)b * NTOT * 4;
  int tid = threadIdx.x;
  __shared__ float rv[256]; __shared__ int ri[256];
  __shared__ float tops[KTOP]; __shared__ int topi[KTOP];
  __shared__ float tb[KTOP][4]; __shared__ int keep[KTOP];

  for (int i = tid; i < NTOT; i += 256) {
    float v = sc[i];
    sc[i] = (v > SCORE_T) ? v : -1.0f;
  }
  __threadfence_block();
  __syncthreads();

  for (int it = 0; it < KTOP; ++it) {
    float best = -1e30f; int bidx = 0x7fffffff;
    for (int i = tid; i < NTOT; i += 256) {
      float v = sc[i];
      if (v > best || (v == best && i < bidx)) { best = v; bidx = i; }
    }
    rv[tid] = best; ri[tid] = bidx;
    __syncthreads();
    for (int s = 128; s > 0; s >>= 1) {
      if (tid < s) {
        float ov = rv[tid + s]; int oi = ri[tid + s];
        if (ov > rv[tid] || (ov == rv[tid] && oi < ri[tid])) { rv[tid] = ov; ri[tid] = oi; }
      }
      __syncthreads();
    }
    if (tid == 0) {
      tops[it] = rv[0]; topi[it] = ri[0];
      sc[ri[0]] = -2.0f;
      __threadfence_block();
    }
    __syncthreads();
  }

  if (tid < KTOP) {
    int i = topi[tid];
    tb[tid][0] = bx[(size_t)i * 4 + 0]; tb[tid][1] = bx[(size_t)i * 4 + 1];
    tb[tid][2] = bx[(size_t)i * 4 + 2]; tb[tid][3] = bx[(size_t)i * 4 + 3];
    keep[tid] = (tops[tid] > SCORE_T) ? 1 : 0;
  }
  __syncthreads();

  for (int i = 0; i < KTOP; ++i) {
    if (tid < KTOP && tid > i && keep[i] && keep[tid]) {
      float aa = (tb[i][2] - tb[i][0]) * (tb[i][3] - tb[i][1]);
      float ab = (tb[tid][2] - tb[tid][0]) * (tb[tid][3] - tb[tid][1]);
      float lx = fmaxf(tb[i][0], tb[tid][0]), ly = fmaxf(tb[i][1], tb[tid][1]);
      float rx = fminf(tb[i][2], tb[tid][2]), ry = fminf(tb[i][3], tb[tid][3]);
      float inter = fmaxf(rx - lx, 0.f) * fmaxf(ry - ly, 0.f);
      float iou = inter / (aa + ab - inter + 1e-9f);
      if (iou > NMS_T) keep[tid] = 0;
    }
    __syncthreads();
  }

  if (tid < KTOP) {
    float* o = out + (((size_t)b * (NCLS - 1) + c) * KTOP + tid) * 5;
    if (keep[tid]) {
      o[0] = tb[tid][0]; o[1] = tb[tid][1]; o[2] = tb[tid][2]; o[3] = tb[tid][3]; o[4] = tops[tid];
    } else {
      o[0] = 0.f; o[1] = 0.f; o[2] = 0.f; o[3] = 0.f; o[4] = 0.f;
    }
  }
}

// ---------------------------------------------------------------------------
extern "C" void kernel_launch(void* const* d_in, const int* in_sizes, int n_in,
                              void* d_out, int out_size, void* d_ws, size_t ws_size,
                              hipStream_t stream) {
  (void)in_sizes; (void)n_in; (void)out_size; (void)ws_size;
  const int B = 32;
  const float* c3 = (const float*)d_in[0];
  const float* c4 = (const float*)d_in[1];
  const float* c5 = (const float*)d_in[2];
  auto F = [&](int i) { return (const float*)d_in[i]; };

  float* ws  = (float*)d_ws;
  float* P3  = ws;                    // B*64*64*64 = 8388608
  float* P4  = P3  + 8388608;         // B*64*32*32 = 2097152
  float* P5  = P4  + 2097152;         // B*64*16*16 = 524288
  float* P6  = P5  + 524288;          // B*64*8*8   = 131072
  float* P7  = P6  + 131072;          // B*64*4*4   = 32768
  float* TMP = P7  + 32768;           // max temp   = 8388608
  float* CLS = TMP + 8388608;         // B*NTOT*21  = 16472064
  float* REG = CLS + 16472064;        // B*NTOT*4   = 3137536
  float* BOX = REG + 3137536;         // B*NTOT*4   = 3137536
  float* PRB = BOX + 3137536;         // B*20*NTOT  = 15687680
  float* PRI = PRB + 15687680;        // NTOT*4     = 98048

  auto gemm_grid = [](int Cout, int HW, int B_) {
    int pixG  = ((HW >> 4) + 3) >> 2;
    int waves = ((Cout + 15) / 16) * pixG * B_;
    return (waves + 7) / 8;           // 256 threads = 8 waves / block
  };
  auto ew_grid = [](int n) { return (n + 255) / 256; };

  k_priors<<<ew_grid(NTOT), 256, 0, stream>>>(PRI);

  // lateral 1x1 convs (WMMA GEMMs, weights staged in LDS)
  k_gemm_plain_t<20><<<gemm_grid(64, 4096, B), 256, 0, stream>>>(c3, F(3),  F(4),  P3, 64, 4096, B);
  k_gemm_plain_t<56><<<gemm_grid(64, 1024, B), 256, 0, stream>>>(c4, F(6),  F(7),  P4, 64, 1024, B);
  k_gemm_plain_t<80><<<gemm_grid(64,  256, B), 256, 0, stream>>>(c5, F(9),  F(10), P5, 64,  256, B);

  // FPN top-down
  k_add_up2<<<ew_grid(B * 64 * 32 * 32), 256, 0, stream>>>(P4, P5, 64, 32, 32, B);
  k_add_up2<<<ew_grid(B * 64 * 64 * 64), 256, 0, stream>>>(P3, P4, 64, 64, 64, B);

  // partial convs (in place via temp)
  k_pconv16<<<ew_grid(B * 16 * 64 * 64), 256, 0, stream>>>(P3, F(5),  TMP, 64, 64, B);
  k_copy16 <<<ew_grid(B * 16 * 64 * 64), 256, 0, stream>>>(P3, TMP, 64, 64, B);
  k_pconv16<<<ew_grid(B * 16 * 32 * 32), 256, 0, stream>>>(P4, F(8),  TMP, 32, 32, B);
  k_copy16 <<<ew_grid(B * 16 * 32 * 32), 256, 0, stream>>>(P4, TMP, 32, 32, B);
  k_pconv16<<<ew_grid(B * 16 * 16 * 16), 256, 0, stream>>>(P5, F(11), TMP, 16, 16, B);
  k_copy16 <<<ew_grid(B * 16 * 16 * 16), 256, 0, stream>>>(P5, TMP, 16, 16, B);

  // p6 = dws(P5, stride 2): dw -> pw + BN + ReLU
  k_dwconv<<<ew_grid(B * 64 * 8 * 8), 256, 0, stream>>>(P5, F(12), TMP, 64, 16, 16, 2, B);
  k_gemm_pw_bnrelu<<<gemm_grid(64, 64, B), 256, 0, stream>>>(TMP, F(13), F(14), F(15), F(16), F(17), P6, 64, B);
  // p7 = dws(P6, stride 2)
  k_dwconv<<<ew_grid(B * 64 * 4 * 4), 256, 0, stream>>>(P6, F(18), TMP, 64, 8, 8, 2, B);
  k_gemm_pw_bnrelu<<<gemm_grid(64, 16, B), 256, 0, stream>>>(TMP, F(19), F(20), F(21), F(22), F(23), P7, 16, B);

  // detection heads
  const float* feats[5] = {P3, P4, P5, P6, P7};
  const int    Hs[5]    = {64, 32, 16, 8, 4};
  const int    nas[5]   = {4, 6, 6, 6, 4};
  const int    bases[5] = {0, 16384, 22528, 24064, 24448};
  for (int l = 0; l < 5; ++l) {
    int H = Hs[l], HW = H * H, na = nas[l];
    int i0 = 24 + 6 * l;
    // cls: dw 3x3 then 1x1 (na*21)
    k_dwconv<<<ew_grid(B * 64 * HW), 256, 0, stream>>>(feats[l], F(i0), TMP, 64, H, H, 1, B);
    k_gemm_head_t<128><<<gemm_grid(na * NCLS, HW, B), 256, 0, stream>>>(
        TMP, F(i0 + 1), F(i0 + 2), CLS, na * NCLS, HW, B, bases[l], na, NCLS);
    // reg: dw 3x3 then 1x1 (na*4)
    k_dwconv<<<ew_grid(B * 64 * HW), 256, 0, stream>>>(feats[l], F(i0 + 3), TMP, 64, H, H, 1, B);
    k_gemm_head_t<32><<<gemm_grid(na * 4, HW, B), 256, 0, stream>>>(
        TMP, F(i0 + 4), F(i0 + 5), REG, na * 4, HW, B, bases[l], na, 4);
  }

  // decode, softmax, NMS
  k_decode <<<ew_grid(B * NTOT), 256, 0, stream>>>(REG, PRI, BOX, B);
  k_softmax<<<ew_grid(B * NTOT), 256, 0, stream>>>(CLS, PRB, B);
  k_nms<<<B * (NCLS - 1), 256, 0, stream>>>(BOX, PRB, (float*)d_out);
}